// GraphSAGEModel_80736795230369
// MI455X (gfx1250) — compile-verified
//
#include <hip/hip_runtime.h>
#include <math.h>

// ---------------------------------------------------------------------------
// GraphSAGE (2 layers): gather+mean -> Linear(no bias) -> ReLU -> L2 normalize
//   N=50000 nodes, DEG=32 neighbors, D=128 features, all fp32.
// Strategy: memory-bound workload (gathers are L2-resident: x is 25.6MB vs
// 192MB L2). GEMM is tiny (1.6 GFLOP/layer) -> keep fp32 precision and use the
// CDNA5 fp32 WMMA shape V_WMMA_F32_16X16X4_F32.
// ---------------------------------------------------------------------------

typedef __attribute__((ext_vector_type(2))) float v2f;
typedef __attribute__((ext_vector_type(8))) float v8f;

#define NNODES 50000
#define DEG    32
#define DF     128
#define L2EPS  1e-12f

// ---------------------------------------------------------------------------
// Kernel 1: ragged neighbor gather + mean.  One wave32 per node; each lane
// owns 4 consecutive features (float4). Neighbor index is wave-uniform
// (readfirstlane -> s_load), each neighbor row read is a coalesced 512B burst.
// ---------------------------------------------------------------------------
__global__ __launch_bounds__(256) void sage_gather_mean(
    const float* __restrict__ src,   // [N, D] node features
    const int*   __restrict__ adj,   // [N, DEG]
    float*       __restrict__ dst)   // [N, D] per-node neighbor mean
{
  const int wave = (blockIdx.x * blockDim.x + threadIdx.x) >> 5;
  const int lane = threadIdx.x & 31;
  if (wave >= NNODES) return;
  const int node = __builtin_amdgcn_readfirstlane(wave);  // force SGPR

  float4 acc = make_float4(0.f, 0.f, 0.f, 0.f);
  #pragma unroll 4
  for (int j = 0; j < DEG; ++j) {
    const int idx = adj[node * DEG + j];                  // wave-uniform
    const float4 v =
        *(const float4*)(src + (size_t)idx * DF + lane * 4);
    acc.x += v.x; acc.y += v.y; acc.z += v.z; acc.w += v.w;
  }
  const float s = 1.0f / (float)DEG;
  acc.x *= s; acc.y *= s; acc.z *= s; acc.w *= s;
  *(float4*)(dst + (size_t)node * DF + lane * 4) = acc;
}

// ---------------------------------------------------------------------------
// Kernel 2: h = relu(A @ W^T); out = h / max(||h||_2, eps)  (per row)
//   A: [N,128] means, W: [128,128] torch-style [out,in] -> B[k][n] = W[n][k].
// One wave32 per 16-row tile. W staged in LDS (64KB of the 320KB/WGP).
// 8 output 16x16 tiles accumulated over K=128 with v_wmma_f32_16x16x4_f32.
//
// fp32 WMMA VGPR layouts (ISA 7.12.2):
//   A 16x4 : lane L (l16 = L&15, half = L>>4) holds A[M=l16][k0 + 2*half + {0,1}]
//   B 4x16 : lane L holds B[k0 + 2*half + {0,1}][N=l16]
//   C 16x16: lane L, vgpr v holds C[M = v + 8*half][N = l16]
// ---------------------------------------------------------------------------
__global__ __launch_bounds__(256) void sage_fc_relu_norm(
    const float* __restrict__ A,     // [N, D]
    const float* __restrict__ W,     // [D, D] row-major [out][in]
    float*       __restrict__ out)   // [N, D]
{
  __shared__ float wlds[DF * DF];    // 64 KB: whole weight matrix

  // Cooperative float4 load of W into LDS.
  for (int i = threadIdx.x; i < (DF * DF) / 4; i += blockDim.x)
    ((float4*)wlds)[i] = ((const float4*)W)[i];
  __syncthreads();

  const int lane = threadIdx.x & 31;
  const int tile = blockIdx.x * (blockDim.x >> 5) + (threadIdx.x >> 5);
  if (tile * 16 >= NNODES) return;   // whole-wave guard (EXEC all-1 for WMMA)

  const int half = lane >> 4;        // 0 | 1
  const int l16  = lane & 15;
  const int row0 = tile * 16;
  const float* __restrict__ arow = A + (size_t)(row0 + l16) * DF;

  v8f acc[8];
  #pragma unroll
  for (int t = 0; t < 8; ++t)
    acc[t] = (v8f){0.f, 0.f, 0.f, 0.f, 0.f, 0.f, 0.f, 0.f};

  for (int kk = 0; kk < DF; kk += 4) {
    v2f a;                                        // A fragment (this K slice)
    a.x = arow[kk + 2 * half + 0];
    a.y = arow[kk + 2 * half + 1];
    #pragma unroll
    for (int t = 0; t < 8; ++t) {                 // 8 N-tiles of 16 cols
      const float* wr = wlds + (size_t)(t * 16 + l16) * DF + kk + 2 * half;
      v2f b; b.x = wr[0]; b.y = wr[1];            // B[k][n] = W[n][k]
      acc[t] = __builtin_amdgcn_wmma_f32_16x16x4_f32(
          /*neg_a=*/false, a, /*neg_b=*/false, b,
          /*c_mod=*/(short)0, acc[t], /*reuse_a=*/false, /*reuse_b=*/false);
    }
  }

  // ReLU + per-row sum of squares. VGPR v of this lane belongs to row
  // M = v + 8*half; the row's 128 elements live in vgpr v across the 16 lanes
  // of this half over the 8 tiles.
  float ss[8];
  #pragma unroll
  for (int v = 0; v < 8; ++v) ss[v] = 0.f;
  #pragma unroll
  for (int t = 0; t < 8; ++t) {
    #pragma unroll
    for (int v = 0; v < 8; ++v) {
      float h = acc[t][v];
      h = h > 0.f ? h : 0.f;
      acc[t][v] = h;
      ss[v] += h * h;
    }
  }
  // Butterfly reduction over the 16 lanes of each half (masks 1..8 stay
  // inside a half on wave32).
  #pragma unroll
  for (int m = 1; m < 16; m <<= 1) {
    #pragma unroll
    for (int v = 0; v < 8; ++v)
      ss[v] += __shfl_xor(ss[v], m, 32);
  }

  float scale[8];
  #pragma unroll
  for (int v = 0; v < 8; ++v) {
    const float nrm = sqrtf(ss[v]);
    scale[v] = 1.0f / fmaxf(nrm, L2EPS);
  }

  // Scale + store. Stores coalesce across l16 (16 contiguous floats).
  #pragma unroll
  for (int t = 0; t < 8; ++t) {
    #pragma unroll
    for (int v = 0; v < 8; ++v) {
      const int m = v + 8 * half;
      out[(size_t)(row0 + m) * DF + t * 16 + l16] = acc[t][v] * scale[v];
    }
  }
}

// ---------------------------------------------------------------------------
// Launch: gather -> fc -> gather -> fc, ping-ponging two 25.6MB ws buffers.
// ---------------------------------------------------------------------------
extern "C" void kernel_launch(void* const* d_in, const int* in_sizes, int n_in,
                              void* d_out, int out_size, void* d_ws,
                              size_t ws_size, hipStream_t stream) {
  const float* x   = (const float*)d_in[0];   // [N, D] fp32
  const int*   adj = (const int*)  d_in[1];   // [N, DEG] int32
  const float* W0  = (const float*)d_in[2];   // [D, D] fp32
  const float* W1  = (const float*)d_in[3];   // [D, D] fp32
  float* out = (float*)d_out;                 // [N, D] fp32

  float* buf0 = (float*)d_ws;                       // [N, D]
  float* buf1 = buf0 + (size_t)NNODES * DF;         // [N, D]

  const dim3 blk(256);
  const int gatherGrid = (NNODES * 32 + 255) / 256;         // 1 wave / node
  const int fcGrid     = ((NNODES / 16) + 7) / 8;           // 1 wave / 16 rows

  // Layer 0
  sage_gather_mean<<<gatherGrid, blk, 0, stream>>>(x, adj, buf0);
  sage_fc_relu_norm<<<fcGrid, blk, 0, stream>>>(buf0, W0, buf1);
  // Layer 1 (gathers from layer-0 output)
  sage_gather_mean<<<gatherGrid, blk, 0, stream>>>(buf1, adj, buf0);
  sage_fc_relu_norm<<<fcGrid, blk, 0, stream>>>(buf0, W1, out);
}